// CartesianMACE_26482768347384
// MI455X (gfx1250) — compile-verified
//
#include <hip/hip_runtime.h>
#include <math.h>

typedef __attribute__((ext_vector_type(2))) float v2f;
typedef __attribute__((ext_vector_type(8))) float v8f;

#define N_NODES 20000
#define N_EDGES 640000
#define CCH     16

// ---------------------------------------------------------------------------
// Dead-code-eliminated MACE: only the rank-0 (scalar) channel path can reach
// the [1,2] output, so we compute exactly that path.
//   per layer l:
//     hm0  = h0s @ ab_w[l,0].T                      (WMMA, [N,16]x[16,16])
//     A0[dst,ch] += hm0[src,ch] * remb(d_e, ch)     (1 coalesced atomic each)
//     s = A0 ; msgs0 = (w1 + w2 s + w3 s^2) * A0
//     h0s' = cw[l,0,n] @ h0s[n] + mw[l,0,n] @ msgs0[n]   (per-node WMMA)
//   out[i] = sum_n (sum_ch h0s[n,ch]) * pred_w[i,n] + pred_b[i]
// ---------------------------------------------------------------------------

__global__ void k_init(const float* __restrict__ h0, float* __restrict__ hbuf,
                       float* __restrict__ out, const float* __restrict__ pred_b) {
    long i = (long)blockIdx.x * blockDim.x + threadIdx.x;
    if (i == 0) { out[0] = pred_b[0]; out[1] = pred_b[1]; }
    if (i < (long)N_NODES * CCH) hbuf[i] = h0[i];   // h0 is [N,C,1] == flat [N*16]
}

__global__ void k_zero(float* __restrict__ p, long n) {
    long i = (long)blockIdx.x * blockDim.x + threadIdx.x;
    if (i < n) p[i] = 0.0f;
}

// edge distances are layer-invariant: compute once
__global__ void k_dist(const int* __restrict__ ei, const float* __restrict__ pos,
                       float* __restrict__ dbuf) {
    long e = (long)blockIdx.x * blockDim.x + threadIdx.x;
    if (e >= N_EDGES) return;
    int src = ei[e];
    int dst = ei[N_EDGES + e];
    float rx = pos[src * 3 + 0] - pos[dst * 3 + 0];
    float ry = pos[src * 3 + 1] - pos[dst * 3 + 1];
    float rz = pos[src * 3 + 2] - pos[dst * 3 + 2];
    dbuf[e] = sqrtf(rx * rx + ry * ry + rz * rz);
}

// ---------------------------------------------------------------------------
// hm0[n][c] = sum_j h0s[n][j] * ab_w[l,0][c][j]   via f32 WMMA 16x16x4
// one wave per 16-node tile; A = h0s tile (M=node, K=j), B = ab_w^T (K=j, N=c)
// ---------------------------------------------------------------------------
__global__ void k_mix(const float* __restrict__ hcur, const float* __restrict__ ab_w,
                      float* __restrict__ hm0, int layer) {
    int wave = (int)((blockIdx.x * blockDim.x + threadIdx.x) >> 5);
    int lane = threadIdx.x & 31;
    const int NT = N_NODES / 16;                  // 1250, exact
    if (wave >= NT) return;                       // wave-uniform exit
    int row  = lane & 15;
    int half = lane >> 4;
    int n0   = wave * 16;

    const float* W  = ab_w + (long)(layer * 3) * CCH * CCH;  // rank 0: W[c*16+j]
    const float* wr = W + row * CCH;                         // B col c = lane row
    const float* hr = hcur + (long)(n0 + row) * CCH;         // A row = node

    v8f acc = {};
    #pragma unroll
    for (int q = 0; q < 4; ++q) {
        int j0 = 4 * q + 2 * half;                // A: lanes0-15 K=4q,4q+1; 16-31: +2
        v2f a, b;
        a.x = hr[j0];     a.y = hr[j0 + 1];
        b.x = wr[j0];     b.y = wr[j0 + 1];
        acc = __builtin_amdgcn_wmma_f32_16x16x4_f32(false, a, false, b,
                                                    (short)0, acc, false, false);
    }
    // D[M=node_local][N=c]: VGPR i -> rows i (lanes0-15) and i+8 (lanes16-31)
    #pragma unroll
    for (int i = 0; i < 8; ++i)
        hm0[(long)(n0 + i + 8 * half) * CCH + row] = acc[i];
}

// ---------------------------------------------------------------------------
// edge scatter, rank 0 only: 16 lanes per edge (one channel each).
// A0[dst*16+ch] += hm0[src*16+ch] * sqrt(2)*sin(pi*(ch+1)*d)/d
// the 16 atomics per edge are contiguous -> coalesced atomic burst into L2.
// ---------------------------------------------------------------------------
__global__ void k_edge(const int* __restrict__ ei, const float* __restrict__ dbuf,
                       const float* __restrict__ hm0, float* __restrict__ A0) {
    long t = (long)blockIdx.x * blockDim.x + threadIdx.x;
    long e = t >> 4;
    int ch = (int)(t & 15);
    if (e >= N_EDGES) return;
    int src = ei[e];
    int dst = ei[N_EDGES + e];
    float d    = dbuf[e];
    float remb = 1.41421356237f * __sinf(3.14159265359f * (float)(ch + 1) * d) / d;
    atomicAdd(&A0[(long)dst * CCH + ch], hm0[(long)src * CCH + ch] * remb);
}

// ---------------------------------------------------------------------------
// per-node update: h' = cw[l,0,n] @ h + mw[l,0,n] @ msgs,  msgs = poly(s) * A0
// one wave per node; 8 chained f32 WMMAs; B/D occupy column 0 only.
// bound by streaming 2 KB/node of weights (82 MB total -> ~3.5 us @ 23.3 TB/s)
// ---------------------------------------------------------------------------
__global__ void k_node(const float* __restrict__ hcur, const float* __restrict__ A0,
                       const float* __restrict__ cw, const float* __restrict__ mw,
                       const float* __restrict__ ws_w, float* __restrict__ hnext,
                       int layer) {
    int wv   = threadIdx.x >> 5;
    int lane = threadIdx.x & 31;
    int n    = blockIdx.x * 8 + wv;               // N divisible by 8
    int row  = lane & 15;
    int half = lane >> 4;

    const float* Wc = cw + ((long)(layer * 3) * N_NODES + n) * (CCH * CCH);
    const float* Wm = mw + ((long)(layer * 3) * N_NODES + n) * (CCH * CCH);
    const float* w  = ws_w + (long)(layer * 3) * 3 * CCH;    // [nu][ch], rank 0
    const float* hn = hcur + (long)n * CCH;
    const float* an = A0 + (long)n * CCH;

    v8f acc = {};
    #pragma unroll
    for (int q = 0; q < 4; ++q) {                 // cw @ h   (B col 0 = h)
        int k0 = 4 * q + 2 * half;
        v2f aw, b;
        aw.x = Wc[row * CCH + k0];
        aw.y = Wc[row * CCH + k0 + 1];
        float h0v = hn[k0], h1v = hn[k0 + 1];
        b.x = (row == 0) ? h0v : 0.0f;            // only N-col 0 is real
        b.y = (row == 0) ? h1v : 0.0f;
        acc = __builtin_amdgcn_wmma_f32_16x16x4_f32(false, aw, false, b,
                                                    (short)0, acc, false, false);
    }
    #pragma unroll
    for (int q = 0; q < 4; ++q) {                 // += mw @ msgs
        int k0 = 4 * q + 2 * half;
        v2f aw, b;
        aw.x = Wm[row * CCH + k0];
        aw.y = Wm[row * CCH + k0 + 1];
        float s0 = an[k0], s1 = an[k0 + 1];
        float p0 = w[k0]     + (w[CCH + k0]     + w[2 * CCH + k0]     * s0) * s0;
        float p1 = w[k0 + 1] + (w[CCH + k0 + 1] + w[2 * CCH + k0 + 1] * s1) * s1;
        b.x = (row == 0) ? p0 * s0 : 0.0f;        // msgs[k] = poly(k)*A0[k]
        b.y = (row == 0) ? p1 * s1 : 0.0f;
        acc = __builtin_amdgcn_wmma_f32_16x16x4_f32(false, aw, false, b,
                                                    (short)0, acc, false, false);
    }
    // D col 0 lives in lanes 0 (rows i) and 16 (rows i+8)
    if (row == 0) {
        #pragma unroll
        for (int i = 0; i < 8; ++i)
            hnext[(long)n * CCH + i + 8 * half] = acc[i];
    }
}

// ---------------------------------------------------------------------------
// readout: out[i] += sum_n (sum_ch h[n,ch]) * pred_w[i,n]
// ---------------------------------------------------------------------------
__global__ void k_final(const float* __restrict__ hfin, const float* __restrict__ pred_w,
                        float* __restrict__ out) {
    __shared__ float r0[256], r1[256];
    int tid = threadIdx.x;
    float a0 = 0.0f, a1 = 0.0f;
    for (int n = blockIdx.x * blockDim.x + tid; n < N_NODES; n += gridDim.x * blockDim.x) {
        const float* hp = hfin + (long)n * CCH;
        float t0 = 0.0f;
        #pragma unroll
        for (int ch = 0; ch < CCH; ++ch) t0 += hp[ch];
        a0 += t0 * pred_w[n];
        a1 += t0 * pred_w[N_NODES + n];
    }
    r0[tid] = a0; r1[tid] = a1;
    __syncthreads();
    for (int s = 128; s > 0; s >>= 1) {
        if (tid < s) { r0[tid] += r0[tid + s]; r1[tid] += r1[tid + s]; }
        __syncthreads();
    }
    if (tid == 0) { atomicAdd(&out[0], r0[0]); atomicAdd(&out[1], r1[0]); }
}

// ---------------------------------------------------------------------------
extern "C" void kernel_launch(void* const* d_in, const int* in_sizes, int n_in,
                              void* d_out, int out_size, void* d_ws, size_t ws_size,
                              hipStream_t stream) {
    const float* pos    = (const float*)d_in[0];
    const float* h0     = (const float*)d_in[1];
    const float* ab_w   = (const float*)d_in[2];
    const float* ws_w   = (const float*)d_in[3];
    const float* cw     = (const float*)d_in[4];
    const float* mw     = (const float*)d_in[5];
    const float* pred_w = (const float*)d_in[6];
    const float* pred_b = (const float*)d_in[7];
    const int*   ei     = (const int*)d_in[8];
    float*       out    = (float*)d_out;

    // workspace (floats): dbuf[E] | hm0[N*16] | A0[N*16] | hA[N*16] | hB[N*16]
    float* wsf  = (float*)d_ws;
    float* dbuf = wsf;                                  // E
    float* hm0  = dbuf + (long)N_EDGES;                 // N*16
    float* A0   = hm0 + (long)N_NODES * CCH;            // N*16
    float* hA   = A0  + (long)N_NODES * CCH;            // N*16
    float* hB   = hA  + (long)N_NODES * CCH;            // N*16

    const long nh = (long)N_NODES * CCH;                // 320,000

    k_init<<<(int)((nh + 255) / 256), 256, 0, stream>>>(h0, hA, out, pred_b);
    k_dist<<<(N_EDGES + 255) / 256, 256, 0, stream>>>(ei, pos, dbuf);

    float* cur = hA;
    float* nxt = hB;
    for (int l = 0; l < 2; ++l) {
        k_zero<<<(int)((nh + 255) / 256), 256, 0, stream>>>(A0, nh);

        k_mix<<<(N_NODES / 16 + 3) / 4, 128, 0, stream>>>(cur, ab_w, hm0, l);

        long et = (long)N_EDGES * 16;                   // one lane per (edge,channel)
        k_edge<<<(int)((et + 255) / 256), 256, 0, stream>>>(ei, dbuf, hm0, A0);

        k_node<<<N_NODES / 8, 256, 0, stream>>>(cur, A0, cw, mw, ws_w, nxt, l);

        float* tmp = cur; cur = nxt; nxt = tmp;
    }

    k_final<<<64, 256, 0, stream>>>(cur, pred_w, out);
}